// CTCLossModel_58162447123219
// MI455X (gfx1250) — compile-verified
//
#include <hip/hip_runtime.h>
#include <hip/hip_bf16.h>

// ---------------------------------------------------------------------------
// CTC forward loss, MI455X (gfx1250).
//   B=256 batch, T=2000 time, L=100 labels, C=36 classes, S=2L+1=201 states.
// One wave32 per batch element; 7 DP states per lane (32*7=224 >= 201).
// Logits streamed Global->LDS with the Tensor Data Mover (TDM), double
// buffered, synchronized with s_wait_tensorcnt.  No block barriers needed.
// ---------------------------------------------------------------------------

#define B_SZ   256
#define T_SZ   2000
#define L_SZ   100
#define C_SZ   36
#define S_SZ   (2 * L_SZ + 1)        // 201
#define BLANK  (C_SZ - 1)            // 35
#define WPB    4                     // waves (batch elements) per block
#define TILE_T 50                    // rows per TDM tile
#define NTILES (T_SZ / TILE_T)       // 40 (exact)
#define TILE_F (TILE_T * C_SZ)       // floats per tile buffer (1800)
#define NEGINF (-1.7014118e38f)      // finfo(f32).min / 2, like the reference

typedef unsigned int u32x4 __attribute__((ext_vector_type(4)));
typedef int          i32x4 __attribute__((ext_vector_type(4)));
typedef int          i32x8 __attribute__((ext_vector_type(8)));

// Issue one TDM 2D tile load: TILE_T rows x C_SZ f32 from global -> LDS.
// D# built per CDNA5 ISA §8.3/§8.4 (count=1, type=2 "image", data_size=4B).
__device__ __forceinline__ void tdm_tile_load(unsigned lds_byte_addr,
                                              unsigned long long gaddr) {
  u32x4 g0;
  g0[0] = 1u;                                   // count=1, user descriptor
  g0[1] = lds_byte_addr;                        // lds_addr [63:32]
  g0[2] = (unsigned)gaddr;                      // global_addr low
  g0[3] = ((unsigned)(gaddr >> 32) & 0x01FFFFFFu) | 0x80000000u; // type=2

  i32x8 g1;
  g1[0] = 0x00020000;                           // data_size=2 (4 bytes)
  g1[1] = (C_SZ << 16);                         // tensor_dim0 = 36 (bits 63:48)
  g1[2] = (TILE_T << 16);                       // tensor_dim1 = 50 (bits 95:80)
  g1[3] = (C_SZ << 16);                         // tile_dim0   = 36 (bits 127:112)
  g1[4] = TILE_T;                               // tile_dim1   = 50
  g1[5] = C_SZ;                                 // tensor_dim0_stride = 36
  g1[6] = 0;
  g1[7] = 0;

  i32x4 z4 = {0, 0, 0, 0};                      // groups 2/3 unused (2D tensor)
#if defined(__clang_major__) && (__clang_major__ >= 23)
  i32x8 z8 = {0, 0, 0, 0, 0, 0, 0, 0};
  __builtin_amdgcn_tensor_load_to_lds(g0, g1, z4, z4, z8, 0);
#else
  __builtin_amdgcn_tensor_load_to_lds(g0, g1, z4, z4, 0);
#endif
}

__global__ __launch_bounds__(WPB * 32) void ctc_loss_kernel(
    const float* __restrict__ logits,       // [B,T,C]
    const int*   __restrict__ labels,       // [B,L]
    const int*   __restrict__ input_lengths,  // [B]
    const int*   __restrict__ label_lengths,  // [B]
    float*       __restrict__ out) {        // [B]
  __shared__ float tiles[WPB * 2 * TILE_F]; // 57.6 KB of 320 KB WGP LDS

  const int lane = threadIdx.x & 31;
  const int wid  = __builtin_amdgcn_readfirstlane((int)(threadIdx.x >> 5));
  const int b    = blockIdx.x * WPB + wid;  // one batch element per wave

  const unsigned wbase = (unsigned)wid * (2u * TILE_F);        // float index
  // LDS byte offset of this wave's double buffer (low 32 bits of flat addr
  // of an LDS object == workgroup LDS byte offset, ISA §10.2).
  const unsigned lds_off0 = (unsigned)__builtin_amdgcn_readfirstlane(
      (int)(unsigned)(size_t)(const void*)(tiles + wbase));

  const int in_len  = input_lengths[b];
  const int lab_len = label_lengths[b];

  // --- Per-lane static DP metadata: ext[s], skip[s] for s = 7*lane + j ---
  int  exts[7];
  bool skips[7];
#pragma unroll
  for (int j = 0; j < 7; ++j) {
    const int s = lane * 7 + j;
    int  e  = BLANK;
    bool sk = false;
    if (s < S_SZ && (s & 1)) {
      const int li = (s - 1) >> 1;
      e = labels[b * L_SZ + li];
      if (s >= 3) sk = (e != BLANK) && (e != labels[b * L_SZ + li - 1]);
    }
    exts[j]  = e;
    skips[j] = sk;
  }

  float a[7];
#pragma unroll
  for (int j = 0; j < 7; ++j) a[j] = NEGINF;

  // --- Prime the TDM pipeline: tile 0 -> buffer 0 ---
  const unsigned long long gbase =
      (unsigned long long)(size_t)logits +
      (unsigned long long)b * (unsigned long long)(T_SZ * C_SZ) * 4ull;
  tdm_tile_load(lds_off0, gbase);

  for (int k = 0; k < NTILES; ++k) {
    if (k + 1 < NTILES) {
      tdm_tile_load(lds_off0 + ((unsigned)(k + 1) & 1u) * (TILE_F * 4u),
                    gbase + (unsigned long long)(k + 1) *
                                (unsigned long long)(TILE_F * 4));
      __builtin_amdgcn_s_wait_tensorcnt((short)1);  // oldest tile landed
    } else {
      __builtin_amdgcn_s_wait_tensorcnt((short)0);  // drain
    }

    const unsigned kbase = wbase + ((unsigned)k & 1u) * TILE_F;
#pragma unroll 1
    for (int r = 0; r < TILE_T; ++r) {
      const int t = k * TILE_T + r;
      if (t != 0 && t >= in_len) continue;          // alpha frozen past length
      const unsigned rb = kbase + (unsigned)r * C_SZ;

      // ---- log-softmax normalizer Z over the 36 classes (wave butterfly) --
      const float x0 = tiles[rb + (unsigned)lane];
      const unsigned i2 = (lane < (C_SZ - 32)) ? (unsigned)(lane + 32)
                                               : (unsigned)lane;
      const float x1 = tiles[rb + i2];
      float m = fmaxf(x0, x1);
#pragma unroll
      for (int msk = 16; msk >= 1; msk >>= 1) m = fmaxf(m, __shfl_xor(m, msk, 32));
      float sl = __expf(x0 - m) +
                 ((lane < (C_SZ - 32)) ? __expf(x1 - m) : 0.0f);
#pragma unroll
      for (int msk = 16; msk >= 1; msk >>= 1) sl += __shfl_xor(sl, msk, 32);
      const float Z = m + __logf(sl);

      if (t == 0) {
#pragma unroll
        for (int j = 0; j < 7; ++j) {
          const int s = lane * 7 + j;
          a[j] = (s < 2) ? (tiles[rb + (unsigned)exts[j]] - Z) : NEGINF;
        }
      } else {
        // cross-lane neighbors: previous lane's states s-1, s-2
        float p6 = __shfl_up(a[6], 1, 32);
        float p5 = __shfl_up(a[5], 1, 32);
        if (lane == 0) { p6 = NEGINF; p5 = NEGINF; }
        float na[7];
#pragma unroll
        for (int j = 0; j < 7; ++j) {
          const float am1 = (j == 0) ? p6 : a[j - 1];
          float am2 = (j == 0) ? p5 : ((j == 1) ? p6 : a[j - 2]);
          am2 = skips[j] ? am2 : NEGINF;
          const float mm  = fmaxf(a[j], fmaxf(am1, am2));
          const float acc = mm + __logf(__expf(a[j] - mm) +
                                        __expf(am1 - mm) +
                                        __expf(am2 - mm));
          na[j] = acc + (tiles[rb + (unsigned)exts[j]] - Z);
        }
#pragma unroll
        for (int j = 0; j < 7; ++j) a[j] = na[j];
      }
    }
  }

  // --- loss = -logaddexp(alpha[2*ll], alpha[2*ll-1]) ---
  const int end   = 2 * lab_len;
  const int endm1 = (end - 1 > 0) ? (end - 1) : 0;
  float vl = NEGINF, vp = NEGINF;
#pragma unroll
  for (int j = 0; j < 7; ++j) {
    const int s = lane * 7 + j;
    if (s == end)   vl = a[j];
    if (s == endm1) vp = a[j];
  }
#pragma unroll
  for (int msk = 16; msk >= 1; msk >>= 1) {
    vl = fmaxf(vl, __shfl_xor(vl, msk, 32));
    vp = fmaxf(vp, __shfl_xor(vp, msk, 32));
  }
  if (lane == 0) {
    const float mm = fmaxf(vl, vp);
    out[b] = -(mm + __logf(__expf(vl - mm) + __expf(vp - mm)));
  }
}

extern "C" void kernel_launch(void* const* d_in, const int* in_sizes, int n_in,
                              void* d_out, int out_size, void* d_ws, size_t ws_size,
                              hipStream_t stream) {
  (void)in_sizes; (void)n_in; (void)out_size; (void)d_ws; (void)ws_size;
  const float* logits        = (const float*)d_in[0];
  const int*   labels        = (const int*)d_in[1];
  const int*   input_lengths = (const int*)d_in[2];
  const int*   label_lengths = (const int*)d_in[3];
  float*       out           = (float*)d_out;

  dim3 grid(B_SZ / WPB);      // 64 blocks
  dim3 block(WPB * 32);       // 4 waves per block, one batch element per wave
  ctc_loss_kernel<<<grid, block, 0, stream>>>(logits, labels, input_lengths,
                                              label_lengths, out);
}